// NeuralRNN_39900246180313
// MI455X (gfx1250) — compile-verified
//
#include <hip/hip_runtime.h>
#include <hip/hip_bf16.h>

typedef __attribute__((ext_vector_type(2))) float v2f;
typedef __attribute__((ext_vector_type(8))) float v8f;

#define B_DIM 64
#define T_DIM 2048
#define I_DIM 128
#define H_DIM 128
#define H_PAD 132   // LDS row stride (floats): conflict-free b64 loads across 64 banks

// ---------------------------------------------------------------------------
// Phase 1: xp[b,t,h] = sum_i x[b,t,i]*W_xh[h,i] + b_xh[h]
// One block = 16 rows of (B*T) x full H; 8 waves, wave w -> N-tile w (16 h's).
// Writes xp into the h_seq region of d_out (same shape), consumed by phase 2.
// ---------------------------------------------------------------------------
__global__ __launch_bounds__(256) void rnn_xproj_kernel(
    const float* __restrict__ x, const float* __restrict__ W_xh,
    const float* __restrict__ b_xh, float* __restrict__ xp) {
  const int lane = threadIdx.x & 31;
  const int wave = threadIdx.x >> 5;        // N tile index, 0..7
  const int m    = lane & 15;               // A row / B col within tile
  const int kp   = lane >> 4;               // K pair-half (0/1)
  const long rowBase = (long)blockIdx.x * 16;
  const int  nBase   = wave * 16;

  // A: x row (contiguous K) ; B: W_xh row h=nBase+m (contiguous K=i)
  const float* aPtr = x + (rowBase + m) * I_DIM + kp * 2;
  const float* bPtr = W_xh + (long)(nBase + m) * I_DIM + kp * 2;

  v8f c0 = {};
  v8f c1 = {};
#pragma unroll
  for (int kk = 0; kk < I_DIM / 4; kk += 2) {
    v2f a0 = *(const v2f*)(aPtr + kk * 4);
    v2f b0 = *(const v2f*)(bPtr + kk * 4);
    v2f a1 = *(const v2f*)(aPtr + kk * 4 + 4);
    v2f b1 = *(const v2f*)(bPtr + kk * 4 + 4);
    c0 = __builtin_amdgcn_wmma_f32_16x16x4_f32(false, a0, false, b0,
                                               (short)0, c0, false, false);
    c1 = __builtin_amdgcn_wmma_f32_16x16x4_f32(false, a1, false, b1,
                                               (short)0, c1, false, false);
  }

  const float bias = b_xh[nBase + m];       // lane's N column == m
  // D layout: row M = kp*8 + r, col N = lane&15
  float* outPtr = xp + (rowBase + kp * 8) * H_DIM + nBase + m;
#pragma unroll
  for (int r = 0; r < 8; ++r) {
    outPtr[(long)r * H_DIM] = c0[r] + c1[r] + bias;
  }
}

// ---------------------------------------------------------------------------
// Phase 2: sequential scan. 4 blocks x 16 batch rows. W_hh N-tile held in
// VGPRs per wave for all 2048 steps; h double-buffered in LDS.
// seqbuf holds xp on entry for step t and is overwritten in place with h_t.
// ---------------------------------------------------------------------------
__global__ __launch_bounds__(256) void rnn_scan_kernel(
    const float* __restrict__ W_hh, const float* __restrict__ b_hh,
    float* __restrict__ seqbuf /* xp in, h_seq out */,
    float* __restrict__ h_last) {
  __shared__ float hbuf[2][16][H_PAD];

  const int lane  = threadIdx.x & 31;
  const int wave  = threadIdx.x >> 5;
  const int m     = lane & 15;
  const int kp    = lane >> 4;
  const int nBase = wave * 16;
  const int bBase = blockIdx.x * 16;

  // Pin this wave's B-operand tile of W_hh in registers: 32 x v2f per lane.
  v2f breg[32];
  {
    const float* bPtr = W_hh + (long)(nBase + m) * H_DIM + kp * 2;
#pragma unroll
    for (int kk = 0; kk < 32; ++kk) breg[kk] = *(const v2f*)(bPtr + kk * 4);
  }
  const float bias = b_hh[nBase + m];

  // h_0 = 0
  for (int idx = threadIdx.x; idx < 16 * H_PAD; idx += 256)
    (&hbuf[0][0][0])[idx] = 0.0f;
  __syncthreads();

  const long rstride = (long)T_DIM * H_DIM;  // batch-row stride in seqbuf
  float* ioPtr0 = seqbuf + ((long)(bBase + kp * 8) * T_DIM) * H_DIM + nBase + m;

  for (int t = 0; t < T_DIM; ++t) {
    float* ioPtr = ioPtr0 + (long)t * H_DIM;

    // Issue xp loads early; latency hides under the WMMA chain.
    float xv[8];
#pragma unroll
    for (int r = 0; r < 8; ++r) xv[r] = ioPtr[r * rstride];

    const float* aPtr = &hbuf[t & 1][0][0] + m * H_PAD + kp * 2;
    v8f c0 = {};
    v8f c1 = {};
#pragma unroll
    for (int kk = 0; kk < 32; kk += 2) {
      v2f a0 = *(const v2f*)(aPtr + kk * 4);
      v2f a1 = *(const v2f*)(aPtr + kk * 4 + 4);
      c0 = __builtin_amdgcn_wmma_f32_16x16x4_f32(false, a0, false, breg[kk],
                                                 (short)0, c0, false, false);
      c1 = __builtin_amdgcn_wmma_f32_16x16x4_f32(false, a1, false, breg[kk + 1],
                                                 (short)0, c1, false, false);
    }

    float* dst = &hbuf[(t + 1) & 1][0][0];
#pragma unroll
    for (int r = 0; r < 8; ++r) {
      float v = tanhf(c0[r] + c1[r] + xv[r] + bias);
      ioPtr[r * rstride] = v;                       // h_seq (overwrites xp)
      dst[(kp * 8 + r) * H_PAD + nBase + m] = v;    // next-step state
    }
    __syncthreads();
  }

  // h_last = hbuf[T&1]
  const float* fin = &hbuf[T_DIM & 1][0][0];
  for (int idx = threadIdx.x; idx < 16 * H_DIM; idx += 256) {
    int row = idx / H_DIM, col = idx - row * H_DIM;
    h_last[(long)(bBase + row) * H_DIM + col] = fin[row * H_PAD + col];
  }
}

extern "C" void kernel_launch(void* const* d_in, const int* in_sizes, int n_in,
                              void* d_out, int out_size, void* d_ws, size_t ws_size,
                              hipStream_t stream) {
  (void)in_sizes; (void)n_in; (void)out_size; (void)d_ws; (void)ws_size;
  const float* x    = (const float*)d_in[0];
  const float* W_xh = (const float*)d_in[1];
  const float* b_xh = (const float*)d_in[2];
  const float* W_hh = (const float*)d_in[3];
  const float* b_hh = (const float*)d_in[4];

  float* h_seq  = (float*)d_out;                                  // [B,T,H]
  float* h_last = (float*)d_out + (long)B_DIM * T_DIM * H_DIM;    // [1,B,H]

  // Phase 1: 131072 rows / 16 per block = 8192 blocks.
  rnn_xproj_kernel<<<(B_DIM * T_DIM) / 16, 256, 0, stream>>>(x, W_xh, b_xh, h_seq);

  // Phase 2: 64 batch rows / 16 per block = 4 blocks (independent chains).
  rnn_scan_kernel<<<B_DIM / 16, 256, 0, stream>>>(W_hh, b_hh, h_seq, h_last);
}